// trained_PPCA_12610023981335
// MI455X (gfx1250) — compile-verified
//
#include <hip/hip_runtime.h>
#include <hip/hip_bf16.h>
#include <math.h>

typedef float v2f __attribute__((ext_vector_type(2)));
typedef float v8f __attribute__((ext_vector_type(8)));

#define NFEAT 3072
#define LAT   128
#define BATCH 8192
#define GRAM_SPLIT 8
#define GRAM_KC (NFEAT / GRAM_SPLIT)   // 384

#define ROWS 128        // batch rows per workgroup in main kernel
#define KCH  64         // K chunk for staging
#define RS   68         // Rlds row stride (68 % 64 == 4 -> conflict-free A frags)
#define WP   288        // Wp k-pair stride (288 % 64 == 32 -> halves on disjoint banks)
#define YS   132        // Ylds row stride (132 % 64 == 4)
#define MP   288        // Mp k-pair stride

#define REGA_SZ (ROWS * RS + (KCH / 2) * WP)   // 8704 + 9216 = 17920 (>= 128*132)

__device__ __forceinline__ v8f wmma_f32(v2f a, v2f b, v8f c) {
  // D = A(16x4, f32) * B(4x16, f32) + C(16x16, f32)
  return __builtin_amdgcn_wmma_f32_16x16x4_f32(false, a, false, b, (short)0, c,
                                               false, false);
}

// -------------------------------------------------------------------------
// Kernel 1: Gpart[chunk] = W_chunk^T @ W_chunk   (split-K, 16x16 tiles, WMMA)
// grid = 64 tiles * GRAM_SPLIT chunks, block = 32 (one wave)
// -------------------------------------------------------------------------
__global__ __launch_bounds__(32) void gram_kernel(const float* __restrict__ W,
                                                  float* __restrict__ Gpart) {
  const int blk   = blockIdx.x;
  const int tile  = blk & 63;
  const int chunk = blk >> 6;
  const int i0 = (tile >> 3) * 16;
  const int j0 = (tile & 7) * 16;
  const int l    = threadIdx.x;
  const int m    = l & 15;
  const int koff = (l >> 4) * 2;

  v8f acc = {0.f, 0.f, 0.f, 0.f, 0.f, 0.f, 0.f, 0.f};
  const int kbase = chunk * GRAM_KC;
  for (int k = 0; k < GRAM_KC; k += 4) {
    const int kk = kbase + k + koff;
    v2f a, b;
    a.x = W[(size_t)kk * LAT + i0 + m];
    a.y = W[(size_t)(kk + 1) * LAT + i0 + m];
    b.x = W[(size_t)kk * LAT + j0 + m];
    b.y = W[(size_t)(kk + 1) * LAT + j0 + m];
    acc = wmma_f32(a, b, acc);
  }
  float* dst = Gpart + (size_t)chunk * (LAT * LAT);
#pragma unroll
  for (int e = 0; e < 8; ++e) {
    const int row = i0 + ((l < 16) ? e : e + 8);
    dst[row * LAT + j0 + m] = acc[e];
  }
}

// -------------------------------------------------------------------------
// Kernel 2: M = sum(Gpart) + s2*I ; in-place Gauss-Jordan inverse (SPD, no
// pivoting), logdet(M) from pivots.  1 block, 128 threads (thread = row).
// -------------------------------------------------------------------------
__global__ __launch_bounds__(128) void prep_kernel(const float* __restrict__ Gpart,
                                                   const float* __restrict__ log_var,
                                                   float* __restrict__ Minv,
                                                   float* __restrict__ scalars) {
  __shared__ float Msh[LAT][LAT + 1];
  __shared__ float logdet_sh;
  const int r = threadIdx.x;
  const float s2 = expf(log_var[0]);

  for (int c = 0; c < LAT; ++c) {
    float v = 0.f;
    for (int s = 0; s < GRAM_SPLIT; ++s)
      v += Gpart[(size_t)s * (LAT * LAT) + r * LAT + c];
    if (c == r) v += s2;
    Msh[r][c] = v;
  }
  if (r == 0) logdet_sh = 0.f;
  __syncthreads();

  for (int p = 0; p < LAT; ++p) {
    if (r == p) {
      const float piv = Msh[p][p];
      logdet_sh += logf(piv);
      const float d = 1.0f / piv;
      for (int j = 0; j < LAT; ++j) Msh[p][j] *= d;
      Msh[p][p] = d;
    }
    __syncthreads();
    if (r != p) {
      const float f = Msh[r][p];
      const float d = Msh[p][p];
#pragma unroll 4
      for (int j = 0; j < LAT; ++j) Msh[r][j] -= f * Msh[p][j];
      Msh[r][p] = -f * d;   // fix up column p (in-place GJ inverse)
    }
    __syncthreads();
  }

  for (int c = 0; c < LAT; ++c) Minv[r * LAT + c] = Msh[r][c];
  if (r == 0) scalars[0] = logdet_sh;  // logdet(M)
}

// -------------------------------------------------------------------------
// Kernel 3 helpers: register double-buffered staging.
// rbuf: 8 x float4 of (X - mean); wbuf: 4 k-pairs x 2 float4 of W.
// -------------------------------------------------------------------------
__device__ __forceinline__ void load_chunk(const float* __restrict__ X,
                                           const float* __restrict__ mean,
                                           const float* __restrict__ W,
                                           size_t row0, int kc, int tid,
                                           float4* rbuf, float4* wbuf) {
#pragma unroll
  for (int i = 0; i < 8; ++i) {
    const int f4  = tid + i * 256;            // 0..2047
    const int row = f4 >> 4;                  // (tid>>4) + 16*i
    const int c4  = (f4 & 15) * 4;
    const float4 xv = *(const float4*)(X + (row0 + row) * NFEAT + kc + c4);
    const float4 mv = *(const float4*)(mean + kc + c4);
    float4 rv;
    rv.x = xv.x - mv.x; rv.y = xv.y - mv.y;
    rv.z = xv.z - mv.z; rv.w = xv.w - mv.w;
    rbuf[i] = rv;
  }
#pragma unroll
  for (int i = 0; i < 4; ++i) {
    const int p   = tid + i * 256;            // 0..1023
    const int kp  = p >> 5;                   // 0..31
    const int col = (p & 31) * 4;
    wbuf[2 * i]     = *(const float4*)(W + (size_t)(kc + 2 * kp) * LAT + col);
    wbuf[2 * i + 1] = *(const float4*)(W + (size_t)(kc + 2 * kp + 1) * LAT + col);
  }
}

__device__ __forceinline__ void store_chunk(int tid, const float4* rbuf,
                                            const float4* wbuf,
                                            float* __restrict__ Rlds,
                                            float* __restrict__ Wp,
                                            float* rssp) {
#pragma unroll
  for (int i = 0; i < 8; ++i) {
    const int f4  = tid + i * 256;
    const int row = f4 >> 4;
    const int c4  = (f4 & 15) * 4;
    const float4 rv = rbuf[i];
    rssp[i] += rv.x * rv.x + rv.y * rv.y + rv.z * rv.z + rv.w * rv.w;
    *(float4*)(&Rlds[row * RS + c4]) = rv;
  }
#pragma unroll
  for (int i = 0; i < 4; ++i) {
    const int p   = tid + i * 256;
    const int kp  = p >> 5;
    const int col = (p & 31) * 4;
    const float4 w0 = wbuf[2 * i];
    const float4 w1 = wbuf[2 * i + 1];
    // k-pair interleave: B fragment (K=k,k+1) becomes one aligned b64 load
    *(float4*)(&Wp[kp * WP + col * 2])     = make_float4(w0.x, w1.x, w0.y, w1.y);
    *(float4*)(&Wp[kp * WP + col * 2 + 4]) = make_float4(w0.z, w1.z, w0.w, w1.w);
  }
}

// -------------------------------------------------------------------------
// Kernel 3: per 128-row block:
//   R = X - mean ; Y = R @ W (WMMA, double-buffered LDS staging)
//   rss = rowsum(R*R) ; Z = Y @ Minv (WMMA) ; qyz = rowsum(Y*Z)
//   out = -0.5*(n*log2pi + (n-k)*log_s2 + logdetM + (rss - qyz)/s2)
// grid = 64, block = 256 (8 waves; wave w owns rows w*16..w*16+15)
// -------------------------------------------------------------------------
__global__ __launch_bounds__(256) void main_kernel(const float* __restrict__ X,
                                                   const float* __restrict__ W,
                                                   const float* __restrict__ log_var,
                                                   const float* __restrict__ mean,
                                                   const float* __restrict__ Minv,
                                                   const float* __restrict__ scalars,
                                                   float* __restrict__ out) {
  __shared__ float regionA[REGA_SZ];          // staging (R + Wp), reused as Ylds
  __shared__ float Mp[(LAT / 2) * MP];        // Minv, k-pair interleaved
  __shared__ float redbuf[ROWS * 16];
  __shared__ float rowrss[ROWS];

  float* const Rlds = regionA;
  float* const Wpl  = regionA + ROWS * RS;
  float* const Ylds = regionA;                // alias, used after staging done

  const int tid  = threadIdx.x;
  const int w    = tid >> 5;
  const int l    = tid & 31;
  const int m    = l & 15;
  const int half = l >> 4;
  const int koff = half * 2;

  const float log_s2  = log_var[0];
  const float inv_s2  = expf(-log_s2);
  const float logdetM = scalars[0];

  // stage Minv into LDS with k-pair interleave
  for (int idx = tid; idx < (LAT / 2) * LAT; idx += 256) {
    const int kp = idx >> 7, col = idx & 127;
    const float a0 = Minv[(2 * kp) * LAT + col];
    const float a1 = Minv[(2 * kp + 1) * LAT + col];
    *(float2*)(&Mp[kp * MP + col * 2]) = make_float2(a0, a1);
  }

  v8f acc[8];
  const v8f vzero = {0.f, 0.f, 0.f, 0.f, 0.f, 0.f, 0.f, 0.f};
#pragma unroll
  for (int c = 0; c < 8; ++c) acc[c] = vzero;
  float rssp[8];
#pragma unroll
  for (int i = 0; i < 8; ++i) rssp[i] = 0.f;

  const size_t row0 = (size_t)blockIdx.x * ROWS;
  const int arow = (w * 16 + m) * RS;         // A-fragment row base in Rlds

  float4 rbuf[8], wbuf[8];
  load_chunk(X, mean, W, row0, 0, tid, rbuf, wbuf);

  // ---------------- Y = (X - mean) @ W over K = 3072 -----------------
  for (int kc = 0; kc < NFEAT; kc += KCH) {
    __syncthreads();                 // previous chunk's LDS reads complete
    store_chunk(tid, rbuf, wbuf, Rlds, Wpl, rssp);
    __syncthreads();                 // staging visible
    if (kc + KCH < NFEAT)            // prefetch next chunk into registers
      load_chunk(X, mean, W, row0, kc + KCH, tid, rbuf, wbuf);
    // WMMA over current chunk (loads in flight underneath)
#pragma unroll 4
    for (int kk = 0; kk < KCH; kk += 4) {
      const v2f a = *(const v2f*)(&Rlds[arow + kk + koff]);
      const float* bp = &Wpl[((kk >> 1) + half) * WP + m * 2];
#pragma unroll
      for (int c = 0; c < 8; ++c) {
        const v2f b = *(const v2f*)(bp + c * 32);
        acc[c] = wmma_f32(a, b, acc[c]);
      }
    }
  }

  __syncthreads();   // all WMMA LDS reads done before regionA is reused as Ylds

  // spill Y to LDS (D layout: vgpr e -> row e or e+8 by lane half)
#pragma unroll
  for (int c = 0; c < 8; ++c)
#pragma unroll
    for (int e = 0; e < 8; ++e) {
      const int row = w * 16 + (half ? e + 8 : e);
      Ylds[row * YS + c * 16 + m] = acc[c][e];
    }
  // rss partials -> redbuf
#pragma unroll
  for (int i = 0; i < 8; ++i)
    redbuf[((tid >> 4) + 16 * i) * 16 + (tid & 15)] = rssp[i];
  __syncthreads();
  if (tid < ROWS) {
    float s = 0.f;
#pragma unroll
    for (int j = 0; j < 16; ++j) s += redbuf[tid * 16 + j];
    rowrss[tid] = s;
  }
#pragma unroll
  for (int c = 0; c < 8; ++c) acc[c] = vzero;
  __syncthreads();   // Y visible; rss reduce done (redbuf reusable)

  // ---------------- Z = Y @ Minv, K = 128 -----------------
  const int yrow = (w * 16 + m) * YS;
#pragma unroll 4
  for (int kk = 0; kk < LAT; kk += 4) {
    const v2f a = *(const v2f*)(&Ylds[yrow + kk + koff]);
    const float* bp = &Mp[((kk >> 1) + half) * MP + m * 2];
#pragma unroll
    for (int c = 0; c < 8; ++c) {
      const v2f b = *(const v2f*)(bp + c * 32);
      acc[c] = wmma_f32(a, b, acc[c]);
    }
  }

  // qyz partials: sum_c Z[row][c*16+m] * Y[row][c*16+m]
#pragma unroll
  for (int e = 0; e < 8; ++e) {
    const int row = w * 16 + (half ? e + 8 : e);
    float s = 0.f;
#pragma unroll
    for (int c = 0; c < 8; ++c) s += acc[c][e] * Ylds[row * YS + c * 16 + m];
    redbuf[row * 16 + m] = s;
  }
  __syncthreads();

  if (tid < ROWS) {
    float q = 0.f;
#pragma unroll
    for (int j = 0; j < 16; ++j) q += redbuf[tid * 16 + j];
    const float quad = (rowrss[tid] - q) * inv_s2;
    const float log2pi = 1.837877066409345f;
    const float logdetC = (float)(NFEAT - LAT) * log_s2 + logdetM;
    out[row0 + tid] = -0.5f * ((float)NFEAT * log2pi + logdetC + quad);
  }
}

extern "C" void kernel_launch(void* const* d_in, const int* in_sizes, int n_in,
                              void* d_out, int out_size, void* d_ws, size_t ws_size,
                              hipStream_t stream) {
  const float* X      = (const float*)d_in[0];  // example [8192,3072]
  const float* W      = (const float*)d_in[1];  // W [3072,128]
  const float* logvar = (const float*)d_in[2];  // [1]
  const float* mean   = (const float*)d_in[3];  // [3072]
  float* out = (float*)d_out;

  float* Gpart   = (float*)d_ws;                            // GRAM_SPLIT*128*128
  float* Minv    = Gpart + (size_t)GRAM_SPLIT * LAT * LAT;  // 128*128
  float* scalars = Minv + (size_t)LAT * LAT;                // [0]=logdetM

  gram_kernel<<<dim3(64 * GRAM_SPLIT), dim3(32), 0, stream>>>(W, Gpart);
  prep_kernel<<<dim3(1), dim3(128), 0, stream>>>(Gpart, logvar, Minv, scalars);
  main_kernel<<<dim3(BATCH / ROWS), dim3(256), 0, stream>>>(X, W, logvar, mean,
                                                            Minv, scalars, out);
}